// RegularizedFMNetOptimized_15805479649400
// MI455X (gfx1250) — compile-verified
//
#include <hip/hip_runtime.h>

typedef __attribute__((ext_vector_type(2))) float v2f;
typedef __attribute__((ext_vector_type(8))) float v8f;
typedef __attribute__((ext_vector_type(4))) int   v4i;

#define BB     4
#define NN     50000
#define CC     256
#define KK     128
#define SPLIT  8
#define CHUNK  64
#define NSEG   ((NN + SPLIT - 1) / SPLIT)   // 6250 = 97*64 + 42
#define LMBDA  100.0f
#define EPITCH (CHUNK + 2)                  // LDS pad: dodge 64-bank stride conflicts

typedef __attribute__((address_space(1))) v4i as1_v4i;
typedef __attribute__((address_space(3))) v4i as3_v4i;

// issue one 16-byte async global->LDS transfer for this lane (ASYNCcnt path)
__device__ __forceinline__ void async_b128(const float* gsrc, float* ldst) {
    __builtin_amdgcn_global_load_async_to_lds_b128(
        (as1_v4i*)gsrc, (as3_v4i*)ldst, 0, 0);
}

// ---------------------------------------------------------------------------
// Kernel 1: A[b]  = Ex[b] (K x N) * Fx[b] (N x C)   (mat = 0)
//           Bm[b] = Ey[b] (K x N) * Fy[b] (N x C)   (mat = 1)
// Block = 8 waves; each wave owns one 16x16 tile of the 16(K) x 128(C) strip,
// sharing the E tile staged in LDS. E staging uses async-to-LDS with double
// buffering so the DMA of chunk i+1 overlaps the 16 WMMAs of chunk i.
// ---------------------------------------------------------------------------
__global__ __launch_bounds__(256) void proj_kernel(
    const float* __restrict__ feat_x, const float* __restrict__ feat_y,
    const float* __restrict__ ex,     const float* __restrict__ ey,
    float* __restrict__ partials)
{
    __shared__ float Etile[2][16][EPITCH];

    const int tid  = threadIdx.x;
    const int lane = tid & 31;
    const int wave = tid >> 5;
    const int lo   = lane & 15;
    const int hi   = lane >> 4;

    const int c_tile = blockIdx.x * 8 + wave;   // 0..15
    const int k_tile = blockIdx.y;              // 0..7
    int z = blockIdx.z;                         // 0 .. 2*BB*SPLIT-1
    const int s   = z % SPLIT;  z /= SPLIT;
    const int b   = z % BB;     z /= BB;
    const int mat = z;                          // 0 -> x, 1 -> y

    const float* E = (mat == 0 ? ex : ey) + (size_t)b * KK * NN;
    const float* F = (mat == 0 ? feat_x : feat_y) + (size_t)b * NN * CC;

    const int n_beg = s * NSEG;
    const int n_end = (n_beg + NSEG < NN) ? (n_beg + NSEG) : NN;

    const int col = c_tile * 16 + lo;           // B-fragment column (C index)

    // staging indices: 256 threads x one b128 = 16 rows x 64 floats
    const int ld_row = tid >> 4;                // 0..15
    const int ld_seg = (tid & 15) * 4;          // float4 slot
    const float* Erow = E + (size_t)(k_tile * 16 + ld_row) * NN;

    v8f acc = {};

    const int n_main_end = n_beg + ((n_end - n_beg) / CHUNK) * CHUNK;
    int buf = 0;

    if (n_beg < n_main_end)                      // prime the pipeline
        async_b128(Erow + n_beg + ld_seg, &Etile[0][ld_row][ld_seg]);

    for (int n0 = n_beg; n0 < n_main_end; n0 += CHUNK) {
        __builtin_amdgcn_s_wait_asynccnt(0);
        __syncthreads();                         // Etile[buf] ready for everyone

        if (n0 + CHUNK < n_main_end)             // DMA next chunk into other buffer
            async_b128(Erow + n0 + CHUNK + ld_seg, &Etile[buf ^ 1][ld_row][ld_seg]);

        __builtin_prefetch(&F[(size_t)(n0 + CHUNK) * CC + col], 0, 1);

        #pragma unroll
        for (int nn = 0; nn < CHUNK; nn += 4) {
            // A-frag (16x4): lane holds {K=2*hi, K=2*hi+1} of row `lo`
            v2f a = *(const v2f*)&Etile[buf][lo][nn + 2 * hi];
            // B-frag (4x16): lane holds rows {2*hi, 2*hi+1} of column `col`
            const size_t nb = (size_t)(n0 + nn + 2 * hi);
            v2f bf;
            bf.x = F[nb * CC + col];
            bf.y = F[(nb + 1) * CC + col];
            acc = __builtin_amdgcn_wmma_f32_16x16x4_f32(
                      false, a, false, bf, (short)0, acc, false, false);
        }
        buf ^= 1;
    }

    // ---- tail (42 elements per segment): branch-free clamp + select ----
    if (n_main_end < n_end) {
        __syncthreads();
        #pragma unroll
        for (int q = 0; q < 4; ++q) {
            const int ng = n_main_end + ld_seg + q;
            const int ngc = (ng < n_end - 1) ? ng : (n_end - 1);
            const float v = Erow[ngc];
            Etile[0][ld_row][ld_seg + q] = (ng < n_end) ? v : 0.0f;
        }
        __syncthreads();
        #pragma unroll
        for (int nn = 0; nn < CHUNK; nn += 4) {
            v2f a = *(const v2f*)&Etile[0][lo][nn + 2 * hi];
            const int nb = n_main_end + nn + 2 * hi;
            const int nb0 = (nb     < n_end - 1) ? nb     : (n_end - 1);
            const int nb1 = (nb + 1 < n_end - 1) ? nb + 1 : (n_end - 1);
            const float f0 = F[(size_t)nb0 * CC + col];
            const float f1 = F[(size_t)nb1 * CC + col];
            v2f bf;
            bf.x = (nb     < n_end) ? f0 : 0.0f;
            bf.y = (nb + 1 < n_end) ? f1 : 0.0f;
            acc = __builtin_amdgcn_wmma_f32_16x16x4_f32(
                      false, a, false, bf, (short)0, acc, false, false);
        }
    }

    // partials layout: [mat][b][s][K][C]; C/D layout: VGPR v -> M = v + 8*hi
    float* out = partials + (((size_t)mat * BB + b) * SPLIT + s) * (KK * CC);
    #pragma unroll
    for (int v = 0; v < 8; ++v) {
        const int r = k_tile * 16 + v + 8 * hi;
        out[(size_t)r * CC + col] = acc[v];
    }
}

// ---------------------------------------------------------------------------
// Kernel 2: reduce the SPLIT partial projections -> full[mat][b][K][C]
// ---------------------------------------------------------------------------
__global__ __launch_bounds__(256) void reduce_partials(
    const float* __restrict__ partials, float* __restrict__ full)
{
    const size_t i   = (size_t)blockIdx.x * blockDim.x + threadIdx.x;
    const size_t per = (size_t)KK * CC;
    const size_t mb  = i / per;
    const size_t kc  = i % per;
    const float* p = partials + (mb * SPLIT) * per + kc;
    float sum = 0.0f;
    #pragma unroll
    for (int s = 0; s < SPLIT; ++s) sum += p[(size_t)s * per];
    full[i] = sum;
}

// ---------------------------------------------------------------------------
// Kernel 3: Gram matrices with fp32 WMMA.
//   mat=0: AAt[b] = A[b] * A[b]^T      mat=1: BAt[b] = Bm[b] * A[b]^T
// Both operands are rows of row-major (K x C) matrices -> b64 loads.
// ---------------------------------------------------------------------------
__global__ __launch_bounds__(32) void gram_kernel(
    const float* __restrict__ full, float* __restrict__ gram)
{
    const int lane = threadIdx.x;
    const int lo = lane & 15, hi = lane >> 4;
    const int n_tile = blockIdx.x;              // output col tile (l)
    const int m_tile = blockIdx.y;              // output row tile (k)
    const int b   = blockIdx.z % BB;
    const int mat = blockIdx.z / BB;

    const float* L = full + (((size_t)mat * BB + b) * KK) * CC;  // A or Bm
    const float* R = full + ((size_t)b * KK) * CC;               // always A

    const int mrow = m_tile * 16 + lo;
    const int ncol = n_tile * 16 + lo;

    v8f acc = {};
    #pragma unroll 4
    for (int c = 0; c < CC; c += 4) {
        v2f a  = *(const v2f*)&L[(size_t)mrow * CC + c + 2 * hi];
        v2f bf = *(const v2f*)&R[(size_t)ncol * CC + c + 2 * hi];
        acc = __builtin_amdgcn_wmma_f32_16x16x4_f32(
                  false, a, false, bf, (short)0, acc, false, false);
    }

    float* out = gram + (((size_t)mat * BB + b) * KK) * KK;
    #pragma unroll
    for (int v = 0; v < 8; ++v)
        out[(size_t)(m_tile * 16 + v + 8 * hi) * KK + ncol] = acc[v];
}

// ---------------------------------------------------------------------------
// Kernel 4: resolvant mask D[b,i,j] (gamma = 0.5 -> sqrtf). One block/batch.
// ---------------------------------------------------------------------------
__global__ __launch_bounds__(128) void mask_kernel(
    const float* __restrict__ evx, const float* __restrict__ evy,
    float* __restrict__ D)
{
    __shared__ float red[128];
    __shared__ float e1s[128], e2s[128];
    const int b = blockIdx.x, t = threadIdx.x;

    red[t] = fmaxf(evx[b * KK + t], evy[b * KK + t]);
    __syncthreads();
    for (int off = 64; off > 0; off >>= 1) {
        if (t < off) red[t] = fmaxf(red[t], red[t + off]);
        __syncthreads();
    }
    const float inv_scale = 1.0f / red[0];
    e1s[t] = sqrtf(evx[b * KK + t] * inv_scale);
    e2s[t] = sqrtf(evy[b * KK + t] * inv_scale);
    __syncthreads();

    for (int idx = t; idx < KK * KK; idx += 128) {
        const int i = idx >> 7, j = idx & 127;
        const float g1 = e1s[j], g2 = e2s[i];
        const float d1 = 1.0f / (g1 * g1 + 1.0f);
        const float d2 = 1.0f / (g2 * g2 + 1.0f);
        const float re = g2 * d2 - g1 * d1;
        const float im = d2 - d1;
        D[((size_t)b * KK + i) * KK + j] = re * re + im * im;
    }
}

// ---------------------------------------------------------------------------
// Kernel 5: 512 independent SPD 128x128 solves (Gaussian elimination, no
// pivoting: Gram + 100*diag(D)). Augmented system in LDS (66 KB — fine on
// CDNA5's 320 KB/WGP). One block per (b, i).
// ---------------------------------------------------------------------------
__global__ __launch_bounds__(128) void solve_kernel(
    const float* __restrict__ gram, const float* __restrict__ D,
    float* __restrict__ out)
{
    extern __shared__ float M[];               // [KK][KK+1]
    const int P = KK + 1;
    const int t = threadIdx.x;
    const int b = blockIdx.x >> 7;
    const int i = blockIdx.x & 127;

    const float* AAt  = gram + ((size_t)b) * KK * KK;
    const float* BAt  = gram + ((size_t)(BB + b)) * KK * KK;
    const float* Drow = D + ((size_t)b * KK + i) * KK;

    for (int k = 0; k < KK; ++k) {
        float v = AAt[(size_t)k * KK + t];
        if (k == t) v += LMBDA * Drow[k];
        M[k * P + t] = v;
    }
    M[t * P + KK] = BAt[(size_t)i * KK + t];
    __syncthreads();

    for (int p = 0; p < KK - 1; ++p) {
        const float pivinv = 1.0f / M[p * P + p];
        const int r = p + 1 + t;
        if (r < KK) {
            const float f = M[r * P + p] * pivinv;
            for (int c = p + 1; c <= KK; ++c)
                M[r * P + c] -= f * M[p * P + c];
        }
        __syncthreads();
    }

    for (int p = KK - 1; p >= 0; --p) {
        if (t == 0) M[p * P + KK] /= M[p * P + p];
        __syncthreads();
        const float xp = M[p * P + KK];
        if (t < p) M[t * P + KK] -= M[t * P + p] * xp;
        __syncthreads();
    }

    out[((size_t)b * KK + i) * KK + t] = M[t * P + KK];
}

// ---------------------------------------------------------------------------
extern "C" void kernel_launch(void* const* d_in, const int* in_sizes, int n_in,
                              void* d_out, int out_size, void* d_ws, size_t ws_size,
                              hipStream_t stream)
{
    const float* feat_x  = (const float*)d_in[0];
    const float* feat_y  = (const float*)d_in[1];
    const float* evals_x = (const float*)d_in[2];
    const float* evals_y = (const float*)d_in[3];
    const float* etx     = (const float*)d_in[4];
    const float* ety     = (const float*)d_in[5];
    float* out = (float*)d_out;

    float* ws = (float*)d_ws;
    size_t off = 0;
    float* partials = ws + off; off += (size_t)2 * BB * SPLIT * KK * CC; // 8 MB
    float* full     = ws + off; off += (size_t)2 * BB * KK * CC;         // 1 MB
    float* gram     = ws + off; off += (size_t)2 * BB * KK * KK;         // 0.5 MB
    float* Dm       = ws + off; off += (size_t)BB * KK * KK;             // 0.25 MB

    proj_kernel<<<dim3(2, 8, 2 * BB * SPLIT), 256, 0, stream>>>(
        feat_x, feat_y, etx, ety, partials);

    reduce_partials<<<dim3((2 * BB * KK * CC) / 256), 256, 0, stream>>>(
        partials, full);

    gram_kernel<<<dim3(8, 8, 2 * BB), 32, 0, stream>>>(full, gram);

    mask_kernel<<<dim3(BB), 128, 0, stream>>>(evals_x, evals_y, Dm);

    const int solve_lds = KK * (KK + 1) * (int)sizeof(float); // 66048 B
    (void)hipFuncSetAttribute((const void*)solve_kernel,
                              hipFuncAttributeMaxDynamicSharedMemorySize,
                              solve_lds);
    solve_kernel<<<dim3(BB * KK), 128, solve_lds, stream>>>(gram, Dm, out);
}